// DMTetGeometry_7696581394982
// MI455X (gfx1250) — compile-verified
//
#include <hip/hip_runtime.h>
#include <cmath>
#include <cstdint>

#define TB 256
#define TET_ITERS 4
#define TET_CHUNK (TB * TET_ITERS)   // 1024 tets per block

struct Deltas { int d[7]; };

__device__ __constant__ int TRI_TAB_D[16][6] = {
    {-1,-1,-1,-1,-1,-1},{1,0,2,-1,-1,-1},{4,0,3,-1,-1,-1},{1,4,2,1,3,4},
    {3,1,5,-1,-1,-1},{2,3,0,2,5,3},{1,4,0,1,5,4},{4,2,5,-1,-1,-1},
    {4,5,2,-1,-1,-1},{4,1,0,4,5,1},{3,2,0,3,5,2},{1,3,5,-1,-1,-1},
    {4,1,2,4,3,1},{3,0,4,-1,-1,-1},{2,0,1,-1,-1,-1},{-1,-1,-1,-1,-1,-1}};
__device__ __constant__ int NUM_TRI_D[16] = {0,1,1,2,1,2,2,1,1,2,2,1,2,1,1,0};
__device__ __constant__ int EDGE_P_D[6] = {0,0,0,1,1,2};
__device__ __constant__ int EDGE_Q_D[6] = {1,2,3,2,3,3};

// ---- CDNA5 async global->LDS copy (gfx1250 ASYNCcnt path) ----
__device__ __forceinline__ void async_g2lds_b128(void* lds_ptr, const void* gptr) {
    unsigned lds_off = (unsigned)(unsigned long long)lds_ptr;  // low 32b of flat shared addr == LDS offset
    unsigned long long ga = (unsigned long long)gptr;
    asm volatile("global_load_async_to_lds_b128 %0, %1, off"
                 :: "v"(lds_off), "v"(ga) : "memory");
}

__device__ __forceinline__ int delta_type(int dlt, const Deltas& D) {
    int ty = -1;
#pragma unroll
    for (int j = 0; j < 7; ++j) if (dlt == D.d[j]) ty = j;
    return ty;
}

// Hillis-Steele inclusive scan of one value per thread; returns exclusive prefix.
__device__ __forceinline__ unsigned block_excl_scan(unsigned v, unsigned* s) {
    const int tid = threadIdx.x;
    s[tid] = v; __syncthreads();
#pragma unroll
    for (int off = 1; off < TB; off <<= 1) {
        unsigned x = (tid >= off) ? s[tid - off] : 0u;
        __syncthreads();
        s[tid] += x;
        __syncthreads();
    }
    unsigned incl = s[tid];
    __syncthreads();
    return incl - v;
}

__global__ void zero_kernel(unsigned* __restrict__ p, long long n) {
    long long i = (long long)blockIdx.x * blockDim.x + threadIdx.x;
    long long stride = (long long)gridDim.x * blockDim.x;
    for (; i < n; i += stride) p[i] = 0u;
}

// Pass 1: stream tets through LDS with a 4-deep async pipeline; classify tets,
// mark unique-edge slots (2-bit flags), per-block m1/m2 counts.
// Each thread reads only the LDS slots it loaded itself, and async loads retire
// in order per wave, so staggered s_wait_asynccnt overlaps load k+1.. with compute k.
__global__ void tet_scan_kernel(const int* __restrict__ tets, const float* __restrict__ sdf,
                                long long F, unsigned* __restrict__ flags,
                                unsigned char* __restrict__ codes,
                                unsigned* __restrict__ m1cnt, unsigned* __restrict__ m2cnt,
                                Deltas D) {
    __shared__ int4 tile[TET_CHUNK];
    __shared__ unsigned red[TB];
    const long long base = (long long)blockIdx.x * TET_CHUNK;

#pragma unroll
    for (int k = 0; k < TET_ITERS; ++k) {
        long long t = base + (long long)k * TB + threadIdx.x;
        long long tc = (t < F) ? t : (F - 1);
        async_g2lds_b128(&tile[k * TB + threadIdx.x], tets + 4 * tc);
    }

    unsigned c1 = 0, c2 = 0;
#pragma unroll
    for (int k = 0; k < TET_ITERS; ++k) {
        switch (k) {   // wait only for chunk k; chunks k+1.. stay in flight
            case 0:  asm volatile("s_wait_asynccnt 0x3" ::: "memory"); break;
            case 1:  asm volatile("s_wait_asynccnt 0x2" ::: "memory"); break;
            case 2:  asm volatile("s_wait_asynccnt 0x1" ::: "memory"); break;
            default: asm volatile("s_wait_asynccnt 0x0" ::: "memory"); break;
        }
        int slot = k * TB + threadIdx.x;
        long long t = base + slot;
        if (t >= F) continue;
        int4 v = tile[slot];
        int vv[4] = {v.x, v.y, v.z, v.w};
        int oc[4];
#pragma unroll
        for (int j = 0; j < 4; ++j) oc[j] = (sdf[vv[j]] > 0.0f) ? 1 : 0;
        int sum = oc[0] + oc[1] + oc[2] + oc[3];
        unsigned char code = 255;
        if (sum > 0 && sum < 4) {
            code = (unsigned char)(oc[0] | (oc[1] << 1) | (oc[2] << 2) | (oc[3] << 3));
            int nt = NUM_TRI_D[code];
            c1 += (nt == 1); c2 += (nt == 2);
#pragma unroll
            for (int eI = 0; eI < 6; ++eI) {
                int a = vv[EDGE_P_D[eI]], b = vv[EDGE_Q_D[eI]];
                int oa = oc[EDGE_P_D[eI]], ob = oc[EDGE_Q_D[eI]];
                int lo = a < b ? a : b, hi = a < b ? b : a;
                int ty = delta_type(hi - lo, D);
                if (ty >= 0) {
                    long long e = (long long)lo * 7 + ty;
                    unsigned bits = 1u | ((oa != ob) ? 2u : 0u);
                    atomicOr(&flags[e >> 4], bits << ((unsigned)(e & 15) * 2u));
                }
            }
        }
        codes[t] = code;
    }
    red[threadIdx.x] = c1; __syncthreads();
    for (int off = TB / 2; off > 0; off >>= 1) {
        if (threadIdx.x < off) red[threadIdx.x] += red[threadIdx.x + off];
        __syncthreads();
    }
    if (threadIdx.x == 0) m1cnt[blockIdx.x] = red[0];
    __syncthreads();
    red[threadIdx.x] = c2; __syncthreads();
    for (int off = TB / 2; off > 0; off >>= 1) {
        if (threadIdx.x < off) red[threadIdx.x] += red[threadIdx.x + off];
        __syncthreads();
    }
    if (threadIdx.x == 0) m2cnt[blockIdx.x] = red[0];
}

// Generic single-block exclusive scan of per-block counts; total -> counters[slot].
__global__ void scan_small_kernel(const unsigned* __restrict__ in, unsigned* __restrict__ out,
                                  int n, int* __restrict__ counters, int slot) {
    __shared__ unsigned s[TB];
    unsigned carry = 0;
    for (int b = 0; b < n; b += TB) {
        int i = b + threadIdx.x;
        unsigned v = (i < n) ? in[i] : 0u;
        unsigned excl = block_excl_scan(v, s);
        if (i < n) out[i] = carry + excl;
        __syncthreads();
        s[threadIdx.x] = v; __syncthreads();
        for (int off = TB / 2; off > 0; off >>= 1) {
            if (threadIdx.x < off) s[threadIdx.x] += s[threadIdx.x + off];
            __syncthreads();
        }
        carry += s[0];
        __syncthreads();
    }
    if (threadIdx.x == 0) counters[slot] = (int)carry;
}

__global__ void finalize_counts_kernel(int* counters) {
    if (threadIdx.x == 0 && blockIdx.x == 0)
        counters[2] = counters[0] + 2 * counters[1];   // Nf = m1 + 2*m2
}

// Pass 2: one flag word (16 edge slots) per thread; popcount the "present & sign" pairs.
__global__ void edge_count_kernel(const unsigned* __restrict__ flags, long long nWords,
                                  unsigned* __restrict__ ecnt) {
    __shared__ unsigned red[TB];
    long long wI = (long long)blockIdx.x * TB + threadIdx.x;
    unsigned c = 0;
    if (wI < nWords) {
        unsigned w = flags[wI];
        c = (unsigned)__popc(w & (w >> 1) & 0x55555555u);
    }
    red[threadIdx.x] = c; __syncthreads();
    for (int off = TB / 2; off > 0; off >>= 1) {
        if (threadIdx.x < off) red[threadIdx.x] += red[threadIdx.x + off];
        __syncthreads();
    }
    if (threadIdx.x == 0) ecnt[blockIdx.x] = red[0];
}

// Pass 3: assign ranks (== np.unique lexicographic order) + interpolate crossing verts.
// One flag word per thread; ranks written back as 4x b128 stores.
__global__ void edge_rank_verts_kernel(const unsigned* __restrict__ flags,
                                       const unsigned* __restrict__ eoff,
                                       const float* __restrict__ pos,
                                       const float* __restrict__ sdf,
                                       long long nWords, int* __restrict__ rankArr,
                                       float* __restrict__ out, Deltas D) {
    __shared__ unsigned s[TB];
    long long wI = (long long)blockIdx.x * TB + threadIdx.x;
    unsigned w = (wI < nWords) ? flags[wI] : 0u;
    unsigned qualMask = w & (w >> 1) & 0x55555555u;
    unsigned cnt = (unsigned)__popc(qualMask);
    unsigned excl = block_excl_scan(cnt, s);
    unsigned r = eoff[blockIdx.x] + excl;
    if (wI >= nWords) return;

    int eBase = (int)(wI * 16);
    int4 rv[4];
    int* rp = (int*)rv;
#pragma unroll
    for (int k = 0; k < 16; ++k) {
        rp[k] = (int)r;
        if ((qualMask >> (2 * k)) & 1u) {
            int e = eBase + k;
            int v0 = e / 7;
            int ty = e - v0 * 7;
            int v1 = v0 + D.d[ty];
            float s0 = sdf[v0], s1 = sdf[v1];
            float den = s0 - s1;
            float w0 = -s1 / den, w1 = s0 / den;
            const float* p0 = pos + 3LL * v0;
            const float* p1 = pos + 3LL * v1;
            float* o = out + 3LL * r;
            o[0] = p0[0] * w0 + p1[0] * w1;
            o[1] = p0[1] * w0 + p1[1] * w1;
            o[2] = p0[2] * w0 + p1[2] * w1;
            r++;
        }
    }
    int4* dst = (int4*)(rankArr + eBase);     // 64B-aligned: eBase % 16 == 0
    dst[0] = rv[0]; dst[1] = rv[1]; dst[2] = rv[2]; dst[3] = rv[3];
}

// Pass 4: emit faces (f1 block then f2 block, stable tet order) and uv_idx.
__global__ void emit_faces_kernel(const int* __restrict__ tets,
                                  const unsigned char* __restrict__ codes,
                                  const int* __restrict__ rankArr,
                                  const unsigned* __restrict__ m1off,
                                  const unsigned* __restrict__ m2off,
                                  const int* __restrict__ counters,
                                  long long F, float* __restrict__ out,
                                  int Nuv, Deltas D) {
    __shared__ unsigned s[TB];
    long long base = (long long)blockIdx.x * TET_CHUNK;
    long long myBase = base + (long long)threadIdx.x * TET_ITERS;   // consecutive ownership
    unsigned char myCode[TET_ITERS];
    unsigned c1 = 0, c2 = 0;
#pragma unroll
    for (int k = 0; k < TET_ITERS; ++k) {
        long long t = myBase + k;
        if (t + 64 < F) __builtin_prefetch(tets + 4 * (t + 64), 0, 0);  // global_prefetch_b8
        unsigned char cd = (t < F) ? codes[t] : (unsigned char)255;
        myCode[k] = cd;
        if (cd != 255) {
            int nt = NUM_TRI_D[cd];
            c1 += (nt == 1); c2 += (nt == 2);
        }
    }
    unsigned excl1 = block_excl_scan(c1, s);
    unsigned excl2 = block_excl_scan(c2, s);
    unsigned r1 = m1off[blockIdx.x] + excl1;
    unsigned r2 = m2off[blockIdx.x] + excl2;
    const int E = counters[3], m1T = counters[0], Nf = counters[2];
    const long long facesBase = 3LL * E;
    const long long uvIdxBase = 3LL * E + 3LL * Nf + 8LL * (long long)Nuv * Nuv;

    for (int k = 0; k < TET_ITERS; ++k) {
        long long t = myBase + k;
        if (t >= F) break;
        unsigned char cd = myCode[k];
        if (cd == 255) continue;
        int nt = NUM_TRI_D[cd];
        int4 v = ((const int4*)tets)[t];
        int vv[4] = {v.x, v.y, v.z, v.w};
        long long rows[2];
        if (nt == 1) { rows[0] = (long long)r1; r1++; }
        else         { rows[0] = (long long)m1T + 2LL * r2; rows[1] = rows[0] + 1; r2++; }
        for (int tri = 0; tri < nt; ++tri) {
            long long r = rows[tri];
#pragma unroll
            for (int j = 0; j < 3; ++j) {
                int slot = TRI_TAB_D[cd][3 * tri + j];
                int a = vv[EDGE_P_D[slot]], b = vv[EDGE_Q_D[slot]];
                int lo = a < b ? a : b, hi = a < b ? b : a;
                int ty = delta_type(hi - lo, D);
                long long e = (long long)lo * 7 + (ty < 0 ? 0 : ty);
                out[facesBase + 3 * r + j] = (float)rankArr[e];
                long long uvv = t * 4 + (j == 0 ? 0 : (tri + j));
                out[uvIdxBase + 3 * r + j] = (float)uvv;
            }
        }
    }
}

// Pass 5: the dominant 403 MB uv atlas. Two b128 stores per cell when the
// device-computed base offset is 16B aligned; scalar fallback otherwise.
__global__ void write_uvs_kernel(float* __restrict__ out, const int* __restrict__ counters,
                                 int Nuv, long long cells) {
    long long idx = (long long)blockIdx.x * blockDim.x + threadIdx.x;
    long long stride = (long long)gridDim.x * blockDim.x;
    const long long base = 3LL * counters[3] + 3LL * counters[2];
    const float inv = 1.0f / (float)Nuv;
    const float pad = 0.9f * inv;
    const bool aligned = ((base & 3) == 0);
    for (; idx < cells; idx += stride) {
        int a = (int)(idx / Nuv), b = (int)(idx % Nuv);
        float tx = b * inv, ty = a * inv;
        float* o = out + base + idx * 8;
        if (aligned) {
            float4* o4 = (float4*)o;
            o4[0] = make_float4(tx, ty, tx + pad, ty);
            o4[1] = make_float4(tx + pad, ty + pad, tx, ty + pad);
        } else {
            o[0] = tx;       o[1] = ty;
            o[2] = tx + pad; o[3] = ty;
            o[4] = tx + pad; o[5] = ty + pad;
            o[6] = tx;       o[7] = ty + pad;
        }
    }
}

extern "C" void kernel_launch(void* const* d_in, const int* in_sizes, int n_in,
                              void* d_out, int out_size, void* d_ws, size_t ws_size,
                              hipStream_t stream) {
    const float* pos = (const float*)d_in[0];
    const float* sdf = (const float*)d_in[1];
    const int* tets  = (const int*)d_in[2];
    const long long Nv = (long long)in_sizes[1];
    const long long F  = (long long)in_sizes[2] / 4;

    int V = (int)floor(cbrt((double)Nv) + 0.5);
    Deltas D;
    D.d[0] = 1;           D.d[1] = V;           D.d[2] = V + 1;
    D.d[3] = V * V;       D.d[4] = V * V + 1;   D.d[5] = V * V + V;
    D.d[6] = V * V + V + 1;                      // ascending == lexicographic (v0,v1)

    long long half = (2 * F + 1) / 2;
    int Nuv = (int)ceil(sqrt((double)half));
    while ((long long)Nuv * Nuv < half) Nuv++;

    const long long NE = 7 * Nv;
    const long long nFlagWords = (NE + 15) / 16;
    const int numTetBlocks  = (int)((F + TET_CHUNK - 1) / TET_CHUNK);
    const int numEdgeBlocks = (int)((nFlagWords + TB - 1) / TB);

    // Carve scratch (256B aligned segments).
    char* w = (char*)d_ws;
    auto carve = [&](size_t bytes) { char* p = w; w += (bytes + 255) & ~(size_t)255; return p; };
    unsigned*      flags   = (unsigned*)carve((size_t)nFlagWords * 4);
    int*           rankArr = (int*)carve((size_t)nFlagWords * 16 * 4);  // padded to word granularity
    unsigned char* codes   = (unsigned char*)carve((size_t)F);
    unsigned*      m1cnt   = (unsigned*)carve((size_t)numTetBlocks * 4);
    unsigned*      m2cnt   = (unsigned*)carve((size_t)numTetBlocks * 4);
    unsigned*      m1off   = (unsigned*)carve((size_t)numTetBlocks * 4);
    unsigned*      m2off   = (unsigned*)carve((size_t)numTetBlocks * 4);
    unsigned*      ecnt    = (unsigned*)carve((size_t)numEdgeBlocks * 4);
    unsigned*      eoff    = (unsigned*)carve((size_t)numEdgeBlocks * 4);
    int*           counters = (int*)carve(64);

    float* out = (float*)d_out;

    zero_kernel<<<4096, TB, 0, stream>>>(flags, nFlagWords);
    tet_scan_kernel<<<numTetBlocks, TB, 0, stream>>>(tets, sdf, F, flags, codes,
                                                     m1cnt, m2cnt, D);
    scan_small_kernel<<<1, TB, 0, stream>>>(m1cnt, m1off, numTetBlocks, counters, 0);
    scan_small_kernel<<<1, TB, 0, stream>>>(m2cnt, m2off, numTetBlocks, counters, 1);
    finalize_counts_kernel<<<1, 1, 0, stream>>>(counters);
    edge_count_kernel<<<numEdgeBlocks, TB, 0, stream>>>(flags, nFlagWords, ecnt);
    scan_small_kernel<<<1, TB, 0, stream>>>(ecnt, eoff, numEdgeBlocks, counters, 3);
    edge_rank_verts_kernel<<<numEdgeBlocks, TB, 0, stream>>>(flags, eoff, pos, sdf, nFlagWords,
                                                             rankArr, out, D);
    emit_faces_kernel<<<numTetBlocks, TB, 0, stream>>>(tets, codes, rankArr, m1off, m2off,
                                                       counters, F, out, Nuv, D);
    long long cells = (long long)Nuv * Nuv;
    int uvBlocks = (int)((cells + TB - 1) / TB);
    if (uvBlocks > 65535) uvBlocks = 65535;
    write_uvs_kernel<<<uvBlocks, TB, 0, stream>>>(out, counters, Nuv, cells);
}